// RoiPooling_3246995275891
// MI455X (gfx1250) — compile-verified
//
#include <hip/hip_runtime.h>
#include <hip/hip_bf16.h>

// ROI max-pool (crop_and_resize 14x14 bilinear + 2x2 maxpool -> 7x7) on
// a [1,50,64,1024] f32 feature map, 1000 ROIs, output [N, 7*7*1024].
//
// MI455X strategy: memory-streaming kernel. Feature map (13.1 MB) is
// L2-resident (192 MB L2); output (200 MB) is written once with
// non-temporal stores so it does not evict the map. One 256-thread
// (8xwave32) block per (roi, py, px); each thread handles 4 channels via
// b128 loads/stores -> 512B fully-coalesced transactions per wave.

namespace {

constexpr int kH = 50;
constexpr int kW = 64;
constexpr int kC = 1024;
constexpr int kPool = 7;   // output 7x7
constexpr int kCS = 14;    // crop size before 2x2 maxpool

typedef float v4f __attribute__((ext_vector_type(4)));

__device__ __forceinline__ v4f v4max(v4f a, v4f b) {
  v4f r;
  r.x = fmaxf(a.x, b.x);
  r.y = fmaxf(a.y, b.y);
  r.z = fmaxf(a.z, b.z);
  r.w = fmaxf(a.w, b.w);
  return r;
}

__global__ __launch_bounds__(256) void roi_maxpool_kernel(
    const float* __restrict__ img,    // [H, W, C]
    const float* __restrict__ rois,   // [N, 4] (y1, x1, y2, x2) in pixels
    const int* __restrict__ stride_p, // scalar extractor stride (16)
    float* __restrict__ out,          // [N, 7*7*C]
    int n_rois) {
  const int px = blockIdx.x;           // 0..6
  const int py = blockIdx.y;           // 0..6
  const int n  = blockIdx.z;           // roi index
  if (n >= n_rois) return;

  // gfx1250 prefetch of the ROI descriptor cacheline (global_prefetch_b8).
  __builtin_prefetch(rois + 4 * n, 0, 3);

  const int c = (int)threadIdx.x * 4;  // 4 channels per thread, 256*4 = 1024

  const float inv_stride = 1.0f / (float)(*stride_p);  // 1/16, exact in fp32

  const v4f r = *(const v4f*)(rois + 4 * n);
  // Match reference op order: rois/stride then /[H,W,H,W] (normalized box).
  const float by1 = (r.x * inv_stride) / (float)kH;
  const float bx1 = (r.y * inv_stride) / (float)kW;
  const float by2 = (r.z * inv_stride) / (float)kH;
  const float bx2 = (r.w * inv_stride) / (float)kW;

  // ys = by1*(H-1) + i * ((by2-by1)*(H-1)/(cs-1)); same for xs.
  const float oy = by1 * (float)(kH - 1);
  const float ox = bx1 * (float)(kW - 1);
  const float dy = (by2 - by1) * (float)(kH - 1) / (float)(kCS - 1);
  const float dx = (bx2 - bx1) * (float)(kW - 1) / (float)(kCS - 1);

  v4f best;
  best.x = -INFINITY; best.y = -INFINITY; best.z = -INFINITY; best.w = -INFINITY;

#pragma unroll
  for (int di = 0; di < 2; ++di) {
    const int i = 2 * py + di;
    const float ys = oy + (float)i * dy;
    const bool vy = (ys >= 0.0f) && (ys <= (float)(kH - 1));
    const float y0f = floorf(ys);
    const float wy = ys - y0f;              // from unclipped floor (ref)
    int yc0 = (int)y0f;
    yc0 = min(max(yc0, 0), kH - 1);
    const int yc1 = min(yc0 + 1, kH - 1);

#pragma unroll
    for (int dj = 0; dj < 2; ++dj) {
      const int j = 2 * px + dj;
      const float xs = ox + (float)j * dx;
      const bool vx = (xs >= 0.0f) && (xs <= (float)(kW - 1));
      const float x0f = floorf(xs);
      const float wx = xs - x0f;
      int xc0 = (int)x0f;
      xc0 = min(max(xc0, 0), kW - 1);
      const int xc1 = min(xc0 + 1, kW - 1);

      v4f v;
      v.x = 0.0f; v.y = 0.0f; v.z = 0.0f; v.w = 0.0f;
      if (vy && vx) {  // uniform across the block (depends on blockIdx only)
        const v4f g00 = *(const v4f*)(img + ((yc0 * kW + xc0) * kC + c));
        const v4f g01 = *(const v4f*)(img + ((yc0 * kW + xc1) * kC + c));
        const v4f g10 = *(const v4f*)(img + ((yc1 * kW + xc0) * kC + c));
        const v4f g11 = *(const v4f*)(img + ((yc1 * kW + xc1) * kC + c));
        const float wx0 = 1.0f - wx;
        const float wy0 = 1.0f - wy;
        // Reference order: top = g00*(1-wx)+g01*wx; bot likewise; blend in y.
        v.x = (g00.x * wx0 + g01.x * wx) * wy0 + (g10.x * wx0 + g11.x * wx) * wy;
        v.y = (g00.y * wx0 + g01.y * wx) * wy0 + (g10.y * wx0 + g11.y * wx) * wy;
        v.z = (g00.z * wx0 + g01.z * wx) * wy0 + (g10.z * wx0 + g11.z * wx) * wy;
        v.w = (g00.w * wx0 + g01.w * wx) * wy0 + (g10.w * wx0 + g11.w * wx) * wy;
      }
      best = v4max(best, v);  // max over the 4 crop cells (zeros if invalid)
    }
  }

  // Streaming NT store: output is write-once (200 MB); keep feature map in L2.
  const size_t o = (((size_t)n * (kPool * kPool)) + (size_t)(py * kPool + px)) *
                       (size_t)kC + (size_t)c;
  __builtin_nontemporal_store(best, (v4f*)(out + o));
}

}  // namespace

extern "C" void kernel_launch(void* const* d_in, const int* in_sizes, int n_in,
                              void* d_out, int out_size, void* d_ws, size_t ws_size,
                              hipStream_t stream) {
  const float* img    = (const float*)d_in[0];   // [1,50,64,1024] f32
  const float* rois   = (const float*)d_in[1];   // [N,4] f32
  const int* stride_p = (const int*)d_in[2];     // scalar int
  float* out          = (float*)d_out;           // [N, 7*7*1024] f32

  const int n_rois = in_sizes[1] / 4;

  dim3 grid(kPool, kPool, (unsigned)n_rois);
  dim3 block(256, 1, 1);  // 8 x wave32; 256 threads * 4ch = 1024 channels
  roi_maxpool_kernel<<<grid, block, 0, stream>>>(img, rois, stride_p, out, n_rois);
}